// VoxelizationByGridShape_1726576856006
// MI455X (gfx1250) — compile-verified
//
#include <hip/hip_runtime.h>
#include <stdint.h>

// ---------------------------------------------------------------------------
// PointPillars hard voxelization for MI455X (gfx1250, wave32).
//
// Pure data-movement workload (no matmul -> no WMMA possible). Design:
//  * dense per-cell state resident in the 192MB L2 (~28MB workspace)
//  * deterministic lock-free top-32 selection via atomicMin insertion cascade
//    (final slot list == 32 smallest point indices, ascending == original
//     point order truncated; order-independent, so graph-replay determinism)
//  * npts derived from slot occupancy via wave32 ballot -> no count atomics
//  * wave32-cooperative emit: 1 wave == 1 voxel == 32 point slots
//  * 128-bit vector loads/stores (global_load_b128 / global_store_b128)
//  * gfx1250 global_prefetch_b8 on the streaming point read
// ---------------------------------------------------------------------------

#define GRID_X   432
#define GRID_Y   496
#define GRID_Z   1
#define NCELLS   (GRID_X * GRID_Y * GRID_Z)   // 214272
#define MAX_PTS  32
#define MAX_VOX  40000
#define INF_U    0xFFFFFFFFu

#define VOX_ELEMS (MAX_VOX * MAX_PTS * 4)     // 5,120,000 floats
#define COOR_OFF  VOX_ELEMS                   // 40000*3 floats of (z,y,x)
#define NPTS_OFF  (VOX_ELEMS + MAX_VOX * 3)   // 40000 floats
#define OUT_TOTAL (NPTS_OFF + MAX_VOX)        // 5,280,000

// Voxel coordinate math bit-matches the JAX reference:
// c = floor((p - pc_min) / voxel_size) in plain IEEE float32, then range-check.
__device__ __forceinline__ bool cell_of(float x, float y, float z, int* cell) {
    float qx = (x - 0.0f)      / 0.16f;
    float qy = (y - (-39.68f)) / 0.16f;
    float qz = (z - (-3.0f))   / 4.0f;
    int cx = (int)floorf(qx);
    int cy = (int)floorf(qy);
    int cz = (int)floorf(qz);
    bool valid = (cx >= 0) & (cx < GRID_X) &
                 (cy >= 0) & (cy < GRID_Y) &
                 (cz >= 0) & (cz < GRID_Z);
    *cell = (cz * GRID_Y + cy) * GRID_X + cx;   // z-major linear id (matches ref)
    return valid;
}

// ---- init workspace: slot arrays = INF, first-point bitmap = 0 ------------
// (wpref region is fully overwritten by k_scan; no init needed)
__global__ void k_init_ws(unsigned* __restrict__ ws, size_t slots_words,
                          size_t total_words) {
    size_t i = (size_t)blockIdx.x * blockDim.x + threadIdx.x;
    size_t stride = (size_t)gridDim.x * blockDim.x;
    for (; i < total_words; i += stride)
        ws[i] = (i < slots_words) ? INF_U : 0u;
}

// ---- init output: voxels=0, coors=-1, npts=0 ------------------------------
__global__ void k_init_out(float* __restrict__ out) {
    size_t i = (size_t)blockIdx.x * blockDim.x + threadIdx.x;
    size_t stride = (size_t)gridDim.x * blockDim.x;
    for (; i < (size_t)OUT_TOTAL; i += stride) {
        bool coor = (i >= (size_t)COOR_OFF) && (i < (size_t)NPTS_OFF);
        out[i] = coor ? -1.0f : 0.0f;
    }
}

// ---- phase 1: lock-free sorted top-32 insertion per cell ------------------
// Cascade invariant: slots only decrease; a value written to slot k is >= the
// final value of slot k-1; any index among the cell's 32 smallest provably
// lands in a slot; no value is duplicated or lost except overflow past slot 31
// (== truncation at MAX_POINTS). Final state is unique for any interleaving.
__global__ void k_insert(const float4* __restrict__ pts, int n,
                         unsigned* __restrict__ slots) {
    int i = blockIdx.x * blockDim.x + threadIdx.x;
    if (i >= n) return;
    if (i + 4096 < n)   // gfx1250 speculative prefetch of the point stream
        __builtin_prefetch((const void*)(pts + i + 4096), 0, 1);
    float4 p = pts[i];                       // global_load_b128
    int cell;
    if (!cell_of(p.x, p.y, p.z, &cell)) return;
    unsigned v = (unsigned)i;
    unsigned* s = slots + (size_t)cell * MAX_PTS;
#pragma unroll 1
    for (int k = 0; k < MAX_PTS; ++k) {
        unsigned old = atomicMin(&s[k], v);
        if (old == INF_U) break;             // absorbed into free slot
        v = (old > v) ? old : v;             // carry the displaced (larger) value
    }                                        // falling off the end == truncation
}

// ---- phase 2: bitmap of "point i is the first point of some voxel" --------
__global__ void k_mark(const unsigned* __restrict__ slots,
                       unsigned* __restrict__ mark, int ncells) {
    int c = blockIdx.x * blockDim.x + threadIdx.x;
    if (c >= ncells) return;
    unsigned f = slots[(size_t)c * MAX_PTS];          // slot0 = min index
    if (f != INF_U) atomicOr(&mark[f >> 5], 1u << (f & 31u));
}

// ---- phase 3: exclusive prefix sum of per-word popcounts (single WG) ------
#define SCAN_T 1024
__global__ void k_scan(const unsigned* __restrict__ mark,
                       unsigned* __restrict__ wpref, int nwords) {
    __shared__ unsigned lds[SCAN_T];
    __shared__ unsigned run_s;
    int tid = threadIdx.x;
    if (tid == 0) run_s = 0u;
    __syncthreads();
    int ntiles = (nwords + SCAN_T - 1) / SCAN_T;
    for (int t = 0; t < ntiles; ++t) {
        int i = t * SCAN_T + tid;
        unsigned v = (i < nwords) ? (unsigned)__popc(mark[i]) : 0u;
        lds[tid] = v;
        __syncthreads();
        for (int off = 1; off < SCAN_T; off <<= 1) {       // Hillis-Steele
            unsigned add = (tid >= off) ? lds[tid - off] : 0u;
            __syncthreads();
            lds[tid] += add;
            __syncthreads();
        }
        unsigned incl = lds[tid];
        unsigned base = run_s;
        if (i < nwords) wpref[i] = base + incl - v;        // exclusive
        __syncthreads();
        if (tid == SCAN_T - 1) run_s = base + incl;
        __syncthreads();
    }
}

// ---- phase 4: one wave32 per cell -> rank, gather, emit -------------------
__global__ void k_emit(const float4* __restrict__ pts,
                       const unsigned* __restrict__ slots,
                       const unsigned* __restrict__ mark,
                       const unsigned* __restrict__ wpref,
                       float* __restrict__ out, int ncells) {
    int gid  = blockIdx.x * blockDim.x + threadIdx.x;
    int cell = gid >> 5;          // wave32: one wave per cell
    int lane = gid & 31;
    if (cell >= ncells) return;   // wave-uniform (all 32 lanes share cell)

    // one coalesced b32 load per wave covers the whole 32-slot list
    unsigned idx = slots[(size_t)cell * MAX_PTS + lane];

    // slot list is sorted ascending with INF tail:
    //   slot0 (broadcast)     = first-occurrence point index
    //   #non-INF slots        = min(count, 32) == npts   (multiset conservation)
    unsigned f  = (unsigned)__builtin_amdgcn_readfirstlane((int)idx);
    if (f == INF_U) return;                          // empty cell
    unsigned np = (unsigned)__popc(__builtin_amdgcn_ballot_w32(idx != INF_U));

    unsigned w = f >> 5, b = f & 31u;
    unsigned rank = wpref[w] + (unsigned)__popc(mark[w] & ((1u << b) - 1u));
    if (rank >= (unsigned)MAX_VOX) return;           // dropped beyond MAX_VOXELS

    float4 val = make_float4(0.f, 0.f, 0.f, 0.f);
    if ((unsigned)lane < np)
        val = pts[idx];                              // random-gather b128
    ((float4*)out)[(size_t)rank * MAX_PTS + lane] = val;   // 128-bit store

    if (lane == 0) {
        int cx = cell % GRID_X;
        int cy = (cell / GRID_X) % GRID_Y;
        int cz = cell / (GRID_X * GRID_Y);
        float* co = out + COOR_OFF + (size_t)rank * 3;
        co[0] = (float)cz;
        co[1] = (float)cy;
        co[2] = (float)cx;
        out[NPTS_OFF + rank] = (float)np;
    }
}

// ---------------------------------------------------------------------------
extern "C" void kernel_launch(void* const* d_in, const int* in_sizes, int n_in,
                              void* d_out, int out_size, void* d_ws, size_t ws_size,
                              hipStream_t stream) {
    const float* points = (const float*)d_in[0];
    const int n = in_sizes[0] / 4;                  // [N,4] float32
    const int nwords = (n + 31) / 32;

    // workspace layout (unsigned words): slots | mark | wpref
    unsigned* slots = (unsigned*)d_ws;                          // NCELLS*32
    size_t slots_words = (size_t)NCELLS * MAX_PTS;
    unsigned* mark  = slots + slots_words;                      // nwords
    unsigned* wpref = mark + nwords;                            // nwords
    size_t init_words = slots_words + (size_t)nwords;           // wpref: no init

    float* out = (float*)d_out;

    k_init_ws <<<2048, 256, 0, stream>>>(slots, slots_words, init_words);
    k_init_out<<<2048, 256, 0, stream>>>(out);
    k_insert  <<<(n + 255) / 256, 256, 0, stream>>>(
        (const float4*)points, n, slots);
    k_mark    <<<(NCELLS + 255) / 256, 256, 0, stream>>>(slots, mark, NCELLS);
    k_scan    <<<1, SCAN_T, 0, stream>>>(mark, wpref, nwords);
    k_emit    <<<((size_t)NCELLS * 32 + 255) / 256, 256, 0, stream>>>(
        (const float4*)points, slots, mark, wpref, out, NCELLS);
}